// WindowsOptim_77111842832477
// MI455X (gfx1250) — compile-verified
//
#include <hip/hip_runtime.h>
#include <hip/hip_bf16.h>
#include <stdint.h>

// Problem geometry (fixed by the reference): (B=8, C=3, H=512, W=768) fp32.
#define W_IMG 768
#define H_IMG 512
#define HALO  5

// Tile: 64x16 outputs per 256-thread block (8 wave32s).
#define TX 64
#define TY 16
#define NTHREADS 256
#define LDS_W      (TX + 2 * HALO)   // 74
#define LDS_H      (TY + 2 * HALO)   // 26
#define LDS_STRIDE (LDS_W + 2)       // 76 (pad)
#define LDS_ELEMS  (LDS_H * LDS_W)   // 1924

__device__ __forceinline__ float fast_exp2(float x) {
#if __has_builtin(__builtin_amdgcn_exp2f)
    return __builtin_amdgcn_exp2f(x);   // single v_exp_f32
#else
    return exp2f(x);
#endif
}

__global__ __launch_bounds__(NTHREADS) void WindowsOptim_gfx1250_kernel(
    const float* __restrict__ move_input,
    const float* __restrict__ target,
    float* __restrict__ out)
{
    __shared__ float tile[LDS_H * LDS_STRIDE];

    const int x0 = blockIdx.x * TX;
    const int y0 = blockIdx.y * TY;
    const size_t plane_off = (size_t)blockIdx.z * (size_t)(H_IMG * W_IMG);
    const float* tgp = target + plane_off;
    const int tid = threadIdx.x;

    const bool interior = (x0 >= HALO) && (x0 + TX + HALO <= W_IMG) &&
                          (y0 >= HALO) && (y0 + TY + HALO <= H_IMG);

    if (interior) {
        // CDNA5 async global->LDS DMA fill (per-lane addresses, ASYNCcnt).
        // LDS address operand is the wave-relative LDS byte offset, which is
        // the low 32 bits of the flat pointer to the __shared__ element.
        const float* gbase = tgp + (size_t)(y0 - HALO) * W_IMG + (x0 - HALO);
        for (int i = tid; i < LDS_ELEMS; i += NTHREADS) {
            const int r = i / LDS_W;
            const int c = i - r * LDS_W;
            const uint32_t lds_off =
                (uint32_t)(uintptr_t)(&tile[r * LDS_STRIDE + c]);
            const uint64_t gaddr =
                (uint64_t)(uintptr_t)(gbase + (size_t)r * W_IMG + c);
            asm volatile("global_load_async_to_lds_b32 %0, %1, off"
                         :: "v"(lds_off), "v"(gaddr)
                         : "memory");
        }
        asm volatile("s_wait_asynccnt 0" ::: "memory");
    } else {
        // Boundary tiles: predicated fill with explicit zero padding
        // (reference zero-pads; OOB taps still contribute exp(-0.5|t0|)).
        for (int i = tid; i < LDS_ELEMS; i += NTHREADS) {
            const int r = i / LDS_W;
            const int c = i - r * LDS_W;
            const int gy = y0 - HALO + r;
            const int gx = x0 - HALO + c;
            float v = 0.0f;
            if ((unsigned)gy < (unsigned)H_IMG && (unsigned)gx < (unsigned)W_IMG)
                v = tgp[(size_t)gy * W_IMG + gx];
            tile[r * LDS_STRIDE + c] = v;
        }
    }
    __syncthreads();

    const int tx = tid & (TX - 1);    // 0..63
    const int ty = tid >> 6;          // 0..3
    const float kExp = -0.72134752044448169f;   // -0.5 * log2(e)

#pragma unroll
    for (int rr = 0; rr < TY / (NTHREADS / TX); ++rr) {   // 4 rows/thread
        const int row = ty + rr * (NTHREADS / TX);
        const size_t gidx =
            plane_off + (size_t)(y0 + row) * W_IMG + (size_t)(x0 + tx);

        const float t0 = tile[(HALO + row) * LDS_STRIDE + (HALO + tx)];
        const float m  = move_input[gidx];

        float cost = 0.0f;
        float wsum = 0.0f;
#pragma unroll
        for (int dy = -HALO; dy <= HALO; dy += 2) {
            // tap(y,x) = target(y-dy, x-dx) -> LDS row (HALO+row-dy)
            const float* rowp =
                &tile[(HALO + row - dy) * LDS_STRIDE + (HALO + tx)];
#pragma unroll
            for (int dx = -HALO; dx <= HALO; dx += 2) {
                const float t  = rowp[-dx];
                const float wi = fast_exp2(fabsf(t - t0) * kExp);
                cost = fmaf(fabsf(t - m), wi, cost);
                wsum += wi;
            }
        }

#if __has_builtin(__builtin_amdgcn_rcpf)
        out[gidx] = cost * __builtin_amdgcn_rcpf(wsum);  // wsum > 0 always
#else
        out[gidx] = cost / wsum;
#endif
    }
}

extern "C" void kernel_launch(void* const* d_in, const int* in_sizes, int n_in,
                              void* d_out, int out_size, void* d_ws, size_t ws_size,
                              hipStream_t stream)
{
    (void)n_in; (void)out_size; (void)d_ws; (void)ws_size;

    const float* move_input = (const float*)d_in[0];
    const float* target     = (const float*)d_in[1];
    float*       out        = (float*)d_out;

    const int planes = in_sizes[0] / (H_IMG * W_IMG);   // B*C = 24
    dim3 grid(W_IMG / TX, H_IMG / TY, planes);          // (12, 32, 24)
    dim3 block(NTHREADS);

    WindowsOptim_gfx1250_kernel<<<grid, block, 0, stream>>>(move_input, target, out);
}